// SEEGTransformer_71373766525215
// MI455X (gfx1250) — compile-verified
//
#include <hip/hip_runtime.h>
#include <math.h>

// Problem constants (from reference)
#define B_   2
#define NS_  2
#define E_   32
#define T_   40
#define F_   37
#define D_   128
#define H_   8
#define HD_  16
#define DFF_ 512
#define L_   6
#define NTOK (B_*NS_*E_*T_)   // 5120 tokens
#define SQ_  (NS_*E_*T_)      // 2560 queries per batch
#define SK_  (E_*T_)          // 1280 keys per batch (ns=0 only)

typedef float v2f __attribute__((ext_vector_type(2)));
typedef float v8f __attribute__((ext_vector_type(8)));
typedef int   vi2 __attribute__((vector_size(2 * sizeof(int))));
typedef int   vi4 __attribute__((vector_size(4 * sizeof(int))));

#define GLOBAL_AS __attribute__((address_space(1)))
#define LDS_AS    __attribute__((address_space(3)))

// D(16x16,f32) = A(16x4,f32) * B(4x16,f32) + C
// A frag (lane l): {A[l%16][koff+0], A[l%16][koff+1]}, koff = 2*(l/16)
// B frag (lane l): {B[koff+0][l%16], B[koff+1][l%16]}
// C/D   (lane l): vgpr v holds element [v + 8*(l/16)][l%16]
static __device__ __forceinline__ v8f wmma4(v2f a, v2f b, v8f c) {
  return __builtin_amdgcn_wmma_f32_16x16x4_f32(false, a, false, b, (short)0, c,
                                               false, false);
}

// ---- CDNA5 async global->LDS copies (ASYNCcnt-tracked DMA into LDS) ----
static __device__ __forceinline__ void async_b64(const float* g, float* l) {
#if __has_builtin(__builtin_amdgcn_global_load_async_to_lds_b64)
  __builtin_amdgcn_global_load_async_to_lds_b64((GLOBAL_AS vi2*)g,
                                                (LDS_AS vi2*)l, 0, 0);
#else
  unsigned lds32 = (unsigned)(size_t)(LDS_AS void*)l;
  unsigned long long ga = (unsigned long long)g;
  asm volatile("global_load_async_to_lds_b64 %0, %1, off"
               :: "v"(lds32), "v"(ga) : "memory");
#endif
}
static __device__ __forceinline__ void async_b128(const float* g, float* l) {
#if __has_builtin(__builtin_amdgcn_global_load_async_to_lds_b128)
  __builtin_amdgcn_global_load_async_to_lds_b128((GLOBAL_AS vi4*)g,
                                                 (LDS_AS vi4*)l, 0, 0);
#else
  unsigned lds32 = (unsigned)(size_t)(LDS_AS void*)l;
  unsigned long long ga = (unsigned long long)g;
  asm volatile("global_load_async_to_lds_b128 %0, %1, off"
               :: "v"(lds32), "v"(ga) : "memory");
#endif
}
static __device__ __forceinline__ void wait_async0() {
#if __has_builtin(__builtin_amdgcn_s_wait_asynccnt)
  __builtin_amdgcn_s_wait_asynccnt(0);
#else
  asm volatile("s_wait_asynccnt 0x0" ::: "memory");
#endif
}

// ---------------- embed: h = x @ Wf + bf + emb[e] ----------------
__global__ void k_embed(const float* __restrict__ x, const float* __restrict__ Wf,
                        const float* __restrict__ bf, const float* __restrict__ emb,
                        float* __restrict__ h) {
  int tok = blockIdx.x;
  int d   = threadIdx.x;
  int e   = (tok / T_) % E_;
  const float* xr = x + (size_t)tok * F_;
  float acc = bf[d] + emb[e * D_ + d];
  for (int f = 0; f < F_; ++f) acc += xr[f] * Wf[f * D_ + d];
  h[(size_t)tok * D_ + d] = acc;
}

// ---------------- layernorm: wave per token, 4 elems/lane ----------------
__global__ void k_ln(const float* __restrict__ h, const float* __restrict__ g,
                     const float* __restrict__ b, float* __restrict__ o) {
  int tok = blockIdx.x, lane = threadIdx.x;
  const float4* hv = (const float4*)(h + (size_t)tok * D_);
  float4 v = hv[lane];
  float s  = v.x + v.y + v.z + v.w;
  float s2 = v.x*v.x + v.y*v.y + v.z*v.z + v.w*v.w;
  for (int m = 16; m >= 1; m >>= 1) { s += __shfl_xor(s, m); s2 += __shfl_xor(s2, m); }
  float mu  = s  * (1.0f / D_);
  float var = s2 * (1.0f / D_) - mu * mu;
  float rs  = rsqrtf(var + 1e-5f);
  int d0 = lane * 4;
  float4 out;
  out.x = (v.x - mu) * rs * g[d0+0] + b[d0+0];
  out.y = (v.y - mu) * rs * g[d0+1] + b[d0+1];
  out.z = (v.z - mu) * rs * g[d0+2] + b[d0+2];
  out.w = (v.w - mu) * rs * g[d0+3] + b[d0+3];
  ((float4*)(o + (size_t)tok * D_))[lane] = out;
}

// ------- WMMA GEMM, async-LDS double-buffered: C = act(A@B + bias) + resid -------
// A: MxK row-major, B: KxN row-major (N,K compile-time).
// Block = 128 threads (4 waves) -> one 16(m) x 64(n) slab; wave w owns n-subtile w.
// Per 16-wide k-step: A tile (16x16) + B tile (16x64) DMA'd to LDS via
// GLOBAL_LOAD_ASYNC_TO_LDS, overlapped with WMMAs on the other buffer.
template<int N, int K, int GELU>
__global__ void k_gemm(const float* __restrict__ A, const float* __restrict__ Bm,
                       const float* __restrict__ bias, const float* __restrict__ resid,
                       float* __restrict__ C) {
  constexpr int KT = K / 16;
  constexpr int AS = 18;   // A LDS row stride (floats): 8B-aligned b64, no bank dup
  constexpr int BS = 72;   // B LDS row stride (floats): 16B-aligned b128 dst
  __shared__ float As[2][16 * AS];
  __shared__ float Bs[2][16 * BS];

  int tid  = threadIdx.x;
  int lane = tid & 31, wid = tid >> 5;
  int m0   = blockIdx.x * 16;
  int nblk = blockIdx.y * 64;
  int mr   = lane & 15;
  int koff = (lane >> 4) * 2;
  int bcol = wid * 16 + mr;

  // cooperative async tile loaders (all 128 threads, EXEC all ones)
  int arow = tid >> 3, acol = (tid & 7) * 2;            // A: 16B-coalesced b64
  int brow0 = (tid * 4) >> 6,        bcol0 = (tid * 4) & 63;          // B: b128
  int brow1 = ((tid + 128) * 4) >> 6, bcol1 = ((tid + 128) * 4) & 63;

  {
    async_b64 (A  + (size_t)(m0 + arow) * K + acol,        &As[0][arow * AS + acol]);
    async_b128(Bm + (size_t)brow0 * N + nblk + bcol0,      &Bs[0][brow0 * BS + bcol0]);
    async_b128(Bm + (size_t)brow1 * N + nblk + bcol1,      &Bs[0][brow1 * BS + bcol1]);
  }

  v8f acc = {0.f,0.f,0.f,0.f,0.f,0.f,0.f,0.f};
  for (int it = 0; it < KT; ++it) {
    int buf = it & 1;
    wait_async0();        // this wave's batch landed in LDS
    __syncthreads();      // everyone's batch landed; previous buffer free
    if (it + 1 < KT) {
      int k0 = (it + 1) * 16;
      async_b64 (A  + (size_t)(m0 + arow) * K + k0 + acol,     &As[buf ^ 1][arow * AS + acol]);
      async_b128(Bm + (size_t)(k0 + brow0) * N + nblk + bcol0, &Bs[buf ^ 1][brow0 * BS + bcol0]);
      async_b128(Bm + (size_t)(k0 + brow1) * N + nblk + bcol1, &Bs[buf ^ 1][brow1 * BS + bcol1]);
    }
    const float* Ar = &As[buf][mr * AS];
#pragma unroll
    for (int c = 0; c < 4; ++c) {
      int kb = c * 4 + koff;
      v2f a = *(const v2f*)(Ar + kb);                 // ds_load_b64
      v2f b;
      b.x = Bs[buf][kb * BS + bcol];
      b.y = Bs[buf][(kb + 1) * BS + bcol];
      acc = wmma4(a, b, acc);
    }
  }

  int half = lane >> 4;
  int n0 = nblk + wid * 16;
#pragma unroll
  for (int v = 0; v < 8; ++v) {
    int row = m0 + v + 8 * half;
    float val = acc[v];
    if (bias) val += bias[n0 + mr];
    if (GELU) val = 0.5f * val * (1.0f + erff(val * 0.70710678118654752f));
    if (resid) val += resid[(size_t)row * N + n0 + mr];
    C[(size_t)row * N + n0 + mr] = val;
  }
}

// ---------------- RoPE + head-major repack ----------------
// mode 0: Q (all ns, rope) -> (B,H,SQ,HD); 1: K (ns0, rope); 2: V (ns0, no rope)
__global__ void k_pack(const float* __restrict__ Y, float* __restrict__ O, int mode) {
  int tid = blockIdx.x * blockDim.x + threadIdx.x;
  int total = (mode == 0 ? NTOK : B_ * SK_) * H_;
  if (tid >= total) return;
  int h = tid & 7, r = tid >> 3;
  int tok, dst, t;
  if (mode == 0) {
    int b = r / SQ_, qg = r % SQ_;
    tok = r; t = r % T_;
    dst = ((b * H_ + h) * SQ_ + qg) * HD_;
  } else {
    int b = r / SK_, kg = r % SK_;
    tok = b * SQ_ + kg; t = kg % T_;
    dst = ((b * H_ + h) * SK_ + kg) * HD_;
  }
  const float* src = Y + (size_t)tok * D_ + h * HD_;
  float vb[16];
#pragma unroll
  for (int i = 0; i < 16; ++i) vb[i] = src[i];
  if (mode < 2) {
    // theta table: idx=[0,2,4,6] -> 160^floor(-idx/32) = {1, 1/160, 1/160, 1/160}
#pragma unroll
    for (int i = 0; i < 4; ++i) {
      float theta = (i == 0) ? 1.0f : (1.0f / 160.0f);
      float ang = (float)t * theta;
      float c = cosf(ang), s = sinf(ang);
      float xl = vb[i], xr = vb[i + 4];
      vb[i]     = xl * c + xr * s;
      vb[i + 4] = xr * c - xl * s;
    }
  }
#pragma unroll
  for (int i = 0; i < 16; ++i) O[dst + i] = vb[i];
}

// ---------------- fused attention (flash-style online softmax) ----------------
__global__ void k_attn(const float* __restrict__ Q, const float* __restrict__ Kk,
                       const float* __restrict__ V, float* __restrict__ AO) {
  int lane = threadIdx.x & 31, wid = threadIdx.x >> 5;
  int qt = blockIdx.x * 4 + wid;      // 0..159
  int hh = blockIdx.y, bb = blockIdx.z;
  int ln16 = lane & 15, half = lane >> 4, koff = half * 2;
  __shared__ float Ps[4][16 * 17];    // padded P tile per wave
  float* P = Ps[wid];

  const float* Qb = Q  + (((size_t)bb * H_ + hh) * SQ_ + qt * 16) * HD_;
  const float* Kb = Kk + (((size_t)bb * H_ + hh) * SK_) * HD_;
  const float* Vb = V  + (((size_t)bb * H_ + hh) * SK_) * HD_;

  v2f qa[4];
  const float* Qr = Qb + ln16 * HD_;
#pragma unroll
  for (int c = 0; c < 4; ++c) qa[c] = *(const v2f*)(Qr + c * 4 + koff);

  int e1[8], t1[8];
#pragma unroll
  for (int v = 0; v < 8; ++v) {
    int qg  = qt * 16 + v + 8 * half;
    int rem = qg % SK_;
    e1[v] = rem / T_;  t1[v] = rem % T_;
  }

  v8f acc = {0.f,0.f,0.f,0.f,0.f,0.f,0.f,0.f};
  float mrun[8], lrun[8];
#pragma unroll
  for (int v = 0; v < 8; ++v) { mrun[v] = -INFINITY; lrun[v] = 0.0f; }

  for (int kt = 0; kt < SK_ / 16; ++kt) {
    v8f s = {0.f,0.f,0.f,0.f,0.f,0.f,0.f,0.f};
    const float* Kr = Kb + (size_t)(kt * 16 + ln16) * HD_;
#pragma unroll
    for (int c = 0; c < 4; ++c) {
      v2f kb = *(const v2f*)(Kr + c * 4 + koff);
      s = wmma4(qa[c], kb, s);
    }
    int key = kt * 16 + ln16;
    int e2 = key / T_, t2 = key % T_;
    float p[8];
#pragma unroll
    for (int v = 0; v < 8; ++v) {
      float sv = s[v] * 0.25f;    // HD^-0.5
      bool msk = (t2 > t1[v]) || ((e2 >= e1[v]) && (t2 == t1[v]));
      sv = msk ? -INFINITY : sv;
      float tmax = sv;
      tmax = fmaxf(tmax, __shfl_xor(tmax, 1));
      tmax = fmaxf(tmax, __shfl_xor(tmax, 2));
      tmax = fmaxf(tmax, __shfl_xor(tmax, 4));
      tmax = fmaxf(tmax, __shfl_xor(tmax, 8));
      float mnew = fmaxf(mrun[v], tmax);
      float sub  = (mnew == -INFINITY) ? 0.0f : mnew;   // ref: m=where(finite,m,0)
      float pv   = __expf(sv - sub);
      float rsum = pv;
      rsum += __shfl_xor(rsum, 1);
      rsum += __shfl_xor(rsum, 2);
      rsum += __shfl_xor(rsum, 4);
      rsum += __shfl_xor(rsum, 8);
      float scale = __expf(mrun[v] - sub);              // -inf -> 0 (acc,l start 0)
      lrun[v] = lrun[v] * scale + rsum;
      acc[v] *= scale;
      mrun[v] = mnew;
      p[v] = pv;
    }
#pragma unroll
    for (int v = 0; v < 8; ++v) P[(v + 8 * half) * 17 + ln16] = p[v];
    asm volatile("s_wait_dscnt 0" ::: "memory");        // cross-lane LDS RAW
    const float* Vr = Vb + (size_t)(kt * 16) * HD_;
    const float* Pr = P + ln16 * 17;
#pragma unroll
    for (int c = 0; c < 4; ++c) {
      v2f pa; pa.x = Pr[c * 4 + koff]; pa.y = Pr[c * 4 + koff + 1];
      v2f vv; vv.x = Vr[(c * 4 + koff) * HD_ + ln16];
              vv.y = Vr[(c * 4 + koff + 1) * HD_ + ln16];
      acc = wmma4(pa, vv, acc);
    }
  }
#pragma unroll
  for (int v = 0; v < 8; ++v) {
    int qg  = qt * 16 + v + 8 * half;
    int tok = bb * SQ_ + qg;
    float den = lrun[v];
    float val = acc[v] / ((den == 0.0f) ? 1.0f : den);  // fully-masked rows -> 0
    AO[(size_t)tok * D_ + hh * HD_ + ln16] = val;
  }
}

// ---------------- final projection: out = h @ Wout + bout ----------------
__global__ void k_out(const float* __restrict__ h, const float* __restrict__ Wout,
                      const float* __restrict__ bout, float* __restrict__ out) {
  int tok = blockIdx.x, lane = threadIdx.x;
  const float4* hv = (const float4*)(h + (size_t)tok * D_);
  const float4* wv = (const float4*)Wout;
  float4 a = hv[lane], w = wv[lane];
  float s = a.x * w.x + a.y * w.y + a.z * w.z + a.w * w.w;
  for (int m = 16; m >= 1; m >>= 1) s += __shfl_xor(s, m);
  if (lane == 0) out[tok] = s + bout[0];
}

extern "C" void kernel_launch(void* const* d_in, const int* in_sizes, int n_in,
                              void* d_out, int out_size, void* d_ws, size_t ws_size,
                              hipStream_t stream) {
  const float* x    = (const float*)d_in[0];
  const float* emb  = (const float*)d_in[1];
  const float* Wf   = (const float*)d_in[2];
  const float* bf   = (const float*)d_in[3];
  const float* Wq   = (const float*)d_in[4];
  const float* bq   = (const float*)d_in[5];
  const float* Wk   = (const float*)d_in[6];
  const float* bk   = (const float*)d_in[7];
  const float* Wv   = (const float*)d_in[8];
  const float* bv   = (const float*)d_in[9];
  const float* Wo   = (const float*)d_in[10];
  const float* bo   = (const float*)d_in[11];
  const float* W1   = (const float*)d_in[12];
  const float* b1   = (const float*)d_in[13];
  const float* W2   = (const float*)d_in[14];
  const float* b2   = (const float*)d_in[15];
  const float* g1   = (const float*)d_in[16];
  const float* be1  = (const float*)d_in[17];
  const float* g2   = (const float*)d_in[18];
  const float* be2  = (const float*)d_in[19];
  const float* Wout = (const float*)d_in[20];
  const float* bout = (const float*)d_in[21];

  float* ws = (float*)d_ws;
  float* h  = ws; ws += (size_t)NTOK * D_;
  float* h2 = ws; ws += (size_t)NTOK * D_;
  float* yq = ws; ws += (size_t)NTOK * D_;
  float* yk = ws; ws += (size_t)NTOK * D_;
  float* yv = ws; ws += (size_t)NTOK * D_;
  float* qp = ws; ws += (size_t)B_ * H_ * SQ_ * HD_;
  float* kp = ws; ws += (size_t)B_ * H_ * SK_ * HD_;
  float* vp = ws; ws += (size_t)B_ * H_ * SK_ * HD_;
  float* ao = ws; ws += (size_t)NTOK * D_;
  float* ff = ws; ws += (size_t)NTOK * DFF_;

  k_embed<<<NTOK, D_, 0, stream>>>(x, Wf, bf, emb, h);

  dim3 gP(NTOK / 16, D_ / 64);     // 320 x 2, 4 waves/block
  dim3 gF(NTOK / 16, DFF_ / 64);   // 320 x 8
  dim3 gA((SQ_ / 16) / 4, H_, B_); // 40 x 8 x 2, 4 waves/block

  for (int l = 0; l < L_; ++l) {
    const float* Wql = Wq + (size_t)l * D_ * D_;
    const float* Wkl = Wk + (size_t)l * D_ * D_;
    const float* Wvl = Wv + (size_t)l * D_ * D_;
    const float* Wol = Wo + (size_t)l * D_ * D_;
    const float* W1l = W1 + (size_t)l * D_ * DFF_;
    const float* W2l = W2 + (size_t)l * DFF_ * D_;
    const float* bql = bq + (size_t)l * D_;
    const float* bkl = bk + (size_t)l * D_;
    const float* bvl = bv + (size_t)l * D_;
    const float* bol = bo + (size_t)l * D_;
    const float* b1l = b1 + (size_t)l * DFF_;
    const float* b2l = b2 + (size_t)l * D_;

    k_ln<<<NTOK, 32, 0, stream>>>(h, g1 + (size_t)l * D_, be1 + (size_t)l * D_, h2);

    k_gemm<D_, D_, 0><<<gP, 128, 0, stream>>>(h2, Wql, bql, nullptr, yq);
    k_gemm<D_, D_, 0><<<gP, 128, 0, stream>>>(h2, Wkl, bkl, nullptr, yk);
    k_gemm<D_, D_, 0><<<gP, 128, 0, stream>>>(h2, Wvl, bvl, nullptr, yv);

    k_pack<<<(NTOK * H_ + 255) / 256, 256, 0, stream>>>(yq, qp, 0);
    k_pack<<<(B_ * SK_ * H_ + 255) / 256, 256, 0, stream>>>(yk, kp, 1);
    k_pack<<<(B_ * SK_ * H_ + 255) / 256, 256, 0, stream>>>(yv, vp, 2);

    k_attn<<<gA, 128, 0, stream>>>(qp, kp, vp, ao);

    // h = h + attn_out @ Wo + bo   (resid read/write is 1:1 per thread)
    k_gemm<D_, D_, 0><<<gP, 128, 0, stream>>>(ao, Wol, bol, h, h);

    k_ln<<<NTOK, 32, 0, stream>>>(h, g2 + (size_t)l * D_, be2 + (size_t)l * D_, h2);
    k_gemm<DFF_, D_, 1><<<gF, 128, 0, stream>>>(h2, W1l, b1l, nullptr, ff); // GELU
    k_gemm<D_, DFF_, 0><<<gP, 128, 0, stream>>>(ff, W2l, b2l, h, h);
  }

  k_out<<<NTOK, 32, 0, stream>>>(h, Wout, bout, (float*)d_out);
}